// DenseCapsuleLayer_34703335752076
// MI455X (gfx1250) — compile-verified
//
#include <hip/hip_runtime.h>
#include <stdint.h>

// Problem constants: B,I,J,M,N = 32,2304,32,8,16 ; NZC = 1152
#define B_DIM  32
#define I_DIM  2304
#define J_DIM  32
#define M_DIM  8
#define N_DIM  16
#define NZC    1152
#define U_ITER 2          // tiles per block; block covers U_ITER*64 k's
#define KSTRIDE 132       // per-k LDS stride in floats (528 B): 128 data + 4 pad

#if defined(__has_builtin)
#if __has_builtin(__builtin_amdgcn_tensor_load_to_lds)
#define USE_TDM 1
#endif
#endif
#ifndef USE_TDM
#define USE_TDM 0
#endif

typedef unsigned int v4u __attribute__((ext_vector_type(4)));
typedef int v4i __attribute__((ext_vector_type(4)));
typedef int v8i __attribute__((ext_vector_type(8)));

// ---------------------------------------------------------------------------
// Kernel 1: per-batch stable compaction of nonzero-row indices.
// wave32 ballot prefix + 8-entry cross-wave scan (3 barriers per chunk).
// ---------------------------------------------------------------------------
__global__ __launch_bounds__(256) void caps_compact(const float* __restrict__ x,
                                                    int* __restrict__ sel) {
  const int b    = blockIdx.x;
  const int tid  = threadIdx.x;
  const int wave = tid >> 5;
  const int lane = tid & 31;
  __shared__ int wtot[8];
  __shared__ int base_s;

  for (int p = tid; p < NZC; p += 256) sel[b * NZC + p] = I_DIM;
  if (tid == 0) base_s = 0;
  __syncthreads();

  const int nchunk = (I_DIM + 255) / 256;  // 9
  for (int c = 0; c < nchunk; ++c) {
    const int r = c * 256 + tid;
    int flag = 0;
    if (r < I_DIM) {
      const float4* px = (const float4*)(x + (size_t)(b * I_DIM + r) * M_DIM);
      float4 a0 = px[0];
      float4 a1 = px[1];
      float s = a0.x + a0.y + a0.z + a0.w + a1.x + a1.y + a1.z + a1.w;
      flag = (s != 0.0f) ? 1 : 0;
    }
    const unsigned bm = (unsigned)__ballot(flag);  // wave32: low 32 bits
    const int prefix  = __popc(bm & ((1u << lane) - 1u));
    if (lane == 0) wtot[wave] = __popc(bm);
    __syncthreads();
    int wbefore = 0, tot = 0;
#pragma unroll
    for (int w = 0; w < 8; ++w) {
      const int t = wtot[w];
      if (w < wave) wbefore += t;
      tot += t;
    }
    if (flag) {
      const int pos = base_s + wbefore + prefix;
      if (pos < NZC) sel[b * NZC + pos] = r;
    }
    __syncthreads();
    if (tid == 0) base_s += tot;
    __syncthreads();
  }
}

// ---------------------------------------------------------------------------
// Kernel 2: u[b,j,k,n] = sum_m W[j,idx_k,n,m] * x[b,idx_k,m]
// Block = 256 thr (8 waves). Wave tile = 8 k's; lane = (k=lane&7, nquad=lane>>3),
// 4 outputs/lane, one b128 store/lane (512 B coalesced per wave store).
// W rows are index-gathered into LDS with one gather-mode TDM instruction per
// tile (TENSORcnt double buffering); async-to-LDS fallback if builtin absent.
// ---------------------------------------------------------------------------
__global__ __launch_bounds__(256) void caps_main(const float* __restrict__ x,
                                                 const float* __restrict__ W,
                                                 const int* __restrict__ sel,
                                                 float* __restrict__ out) {
  // [buf][wave][8 k * KSTRIDE floats]  = 2*8*1056*4 = 67,584 B
  __shared__ __align__(16) float stage[2][8][8 * KSTRIDE];

  const int tid  = threadIdx.x;
  const int wave = tid >> 5;
  const int lane = tid & 31;
  const int myk  = lane & 7;   // k within the wave tile
  const int q    = lane >> 3;  // n-quad (0..3)
  const int b    = blockIdx.z;
  const int j    = blockIdx.y;
  const int* selb = sel + b * NZC;
  const uint64_t wbase = (uint64_t)(uintptr_t)W;

  auto kbase = [&](int u) {
    return (blockIdx.x * U_ITER + u) * 64 + wave * 8;
  };
  auto load_sel = [&](int u) {  // this wave's 8 idx values, one per lane&7
    int idx = selb[kbase(u) + myk];
    if (idx < 0) idx = 0;
    if (idx > I_DIM - 1) idx = I_DIM - 1;  // mirror JAX OOB clamp
    return idx;
  };

#if USE_TDM
  // One gather-mode TDM per tile: gather 8 rows of W[j] (128 f32 each, 16-bit
  // row indices) into LDS with 4-dword padding per row -> KSTRIDE=132 floats.
  auto issue = [&](int u, int selv) {
    const uint64_t gaddr = wbase + (uint64_t)j * (I_DIM * 512u);
    v4u g0;
    g0.x = 0x80000001u;  // gather_mode=1, 16-bit indices, count=1
    g0.y = (unsigned)(uintptr_t)&stage[u & 1][wave][0];       // lds_addr
    g0.z = (unsigned)(gaddr & 0xFFFFFFFFu);                   // global_addr lo
    g0.w = (unsigned)((gaddr >> 32) & 0x01FFFFFFu) | 0x80000000u;  // hi | type=2
    v8i g1;
    g1[0] = (int)((2u << 16) | (1u << 20) | (6u << 22) | (3u << 25));
    // data_size=4B, pad_enable, pad_interval=128 dwords, pad_amount=4 dwords
    g1[1] = (int)(128u << 16);   // tensor_dim0 = 128 (bits[79:48] lo half)
    g1[2] = (int)(2304u << 16);  // tensor_dim1 = 2304 (bits[111:80] lo half)
    g1[3] = (int)(128u << 16);   // tile_dim0 = 128 (bits[127:112])
    g1[4] = 8;                   // tile_dim1 = #valid gather indices
    g1[5] = 128;                 // tensor_dim0_stride = 128 elements
    g1[6] = 0;
    g1[7] = 0;
    const int i0 = __builtin_amdgcn_readlane(selv, 0);
    const int i1 = __builtin_amdgcn_readlane(selv, 1);
    const int i2 = __builtin_amdgcn_readlane(selv, 2);
    const int i3 = __builtin_amdgcn_readlane(selv, 3);
    const int i4 = __builtin_amdgcn_readlane(selv, 4);
    const int i5 = __builtin_amdgcn_readlane(selv, 5);
    const int i6 = __builtin_amdgcn_readlane(selv, 6);
    const int i7 = __builtin_amdgcn_readlane(selv, 7);
    v4i g2;  // row_index_0..7, packed 16-bit
    g2.x = (i0 & 0xFFFF) | (i1 << 16);
    g2.y = (i2 & 0xFFFF) | (i3 << 16);
    g2.z = (i4 & 0xFFFF) | (i5 << 16);
    g2.w = (i6 & 0xFFFF) | (i7 << 16);
    v4i g3 = {0, 0, 0, 0};  // row_index_8..15 unused (tile_dim1 = 8)
#if defined(__clang_major__) && (__clang_major__ >= 23)
    v8i z8 = {0, 0, 0, 0, 0, 0, 0, 0};
    __builtin_amdgcn_tensor_load_to_lds(g0, g1, g2, g3, z8, 0);
#else
    __builtin_amdgcn_tensor_load_to_lds(g0, g1, g2, g3, 0);
#endif
  };
  auto wait_next = [&] { asm volatile("s_wait_tensorcnt 0x1" ::: "memory"); };
  auto wait_last = [&] { asm volatile("s_wait_tensorcnt 0x0" ::: "memory"); };
#else
  // Fallback: 8 async-to-LDS instructions per tile (one 512 B W block each).
  auto issue = [&](int u, int selv) {
    const unsigned sbase =
        (unsigned)(uintptr_t)&stage[u & 1][wave][0] + (unsigned)lane * 16u;
    const unsigned lo16 = (unsigned)lane * 16u;
#pragma unroll
    for (int kk = 0; kk < 8; ++kk) {
      const int idxk = __shfl(selv, kk, 32);
      const unsigned goff =
          (unsigned)((unsigned)(j * I_DIM + idxk) * 512u) + lo16;
      const unsigned loff = sbase + (unsigned)kk * (KSTRIDE * 4u);
      asm volatile("global_load_async_to_lds_b128 %0, %1, %2"
                   :
                   : "v"(loff), "v"(goff), "s"(wbase)
                   : "memory");
    }
  };
  auto wait_next = [&] { asm volatile("s_wait_asynccnt 0x8" ::: "memory"); };
  auto wait_last = [&] { asm volatile("s_wait_asynccnt 0x0" ::: "memory"); };
#endif

  auto compute = [&](int u, int selv) {
    const int idxk = __shfl(selv, myk, 32);
    const float4* xp = (const float4*)(x + (size_t)(b * I_DIM + idxk) * M_DIM);
    const float4 xa = xp[0];
    const float4 xb = xp[1];
    const float* wk = &stage[u & 1][wave][myk * KSTRIDE];
    float4 o;
    float* oc = (float*)&o;
#pragma unroll
    for (int t = 0; t < 4; ++t) {
      const int n = q * 4 + t;
      const float4 w0 = *(const float4*)(wk + n * 8);
      const float4 w1 = *(const float4*)(wk + n * 8 + 4);
      float acc = w0.x * xa.x;
      acc = fmaf(w0.y, xa.y, acc);
      acc = fmaf(w0.z, xa.z, acc);
      acc = fmaf(w0.w, xa.w, acc);
      acc = fmaf(w1.x, xb.x, acc);
      acc = fmaf(w1.y, xb.y, acc);
      acc = fmaf(w1.z, xb.z, acc);
      acc = fmaf(w1.w, xb.w, acc);
      oc[t] = acc;
    }
    float* orow =
        out + ((size_t)(b * J_DIM + j) * NZC + kbase(u) + myk) * N_DIM + q * 4;
    *(float4*)orow = o;
  };

  int sv0 = load_sel(0);
  issue(0, sv0);
#pragma unroll
  for (int u = 0; u < U_ITER; ++u) {
    int svn = 0;
    if (u + 1 < U_ITER) {
      svn = load_sel(u + 1);
      issue(u + 1, svn);
      wait_next();  // next tile may remain outstanding; this tile is complete
    } else {
      wait_last();
    }
    compute(u, sv0);
    sv0 = svn;
  }
}

// ---------------------------------------------------------------------------
extern "C" void kernel_launch(void* const* d_in, const int* in_sizes, int n_in,
                              void* d_out, int out_size, void* d_ws, size_t ws_size,
                              hipStream_t stream) {
  (void)in_sizes; (void)n_in; (void)out_size; (void)ws_size;
  const float* x = (const float*)d_in[0];
  const float* W = (const float*)d_in[1];  // (1,J,I,N,M) == (J,I,N,M) flat
  float* out = (float*)d_out;
  int* sel = (int*)d_ws;  // B*NZC ints = 147,456 B of workspace

  caps_compact<<<dim3(B_DIM), dim3(256), 0, stream>>>(x, sel);
  // grid.x = NZC / (64 * U_ITER) = 1152/128 = 9
  caps_main<<<dim3(NZC / (64 * U_ITER), J_DIM, B_DIM), dim3(256), 0, stream>>>(
      x, W, sel, out);
}